// GCN_31722628448475
// MI455X (gfx1250) — compile-verified
//
#include <hip/hip_runtime.h>
#include <hip/hip_bf16.h>

// ---------------------------------------------------------------------------
// 2-layer GCN for MI455X (gfx1250, wave32).
//   out = Ahat * relu(Ahat*(X@W1)+b1) @ W2 ... with Ahat = D^-1/2 (A+I) D^-1/2
// Norm factorization: norm_e = dinv[src]*dinv[dst]  =>
//   g[j]   = dinv[j] * (feat[j] @ W)         (gather-side scale, fused in GEMM)
//   acc[i] = g[i] + sum_{j->i} g[j]          (self-loop = accumulator init)
//   out[i] = dinv[i] * acc[i] + b            (scatter-side scale at finalize)
// GEMMs use V_WMMA_F32_16X16X4_F32 (exact fp32, matches reference precision).
// ---------------------------------------------------------------------------

#define IN_C  256
#define HID_C 64
#define OUT_C 40

typedef float v2f __attribute__((ext_vector_type(2)));
typedef float v8f __attribute__((ext_vector_type(8)));

// ---------------------------- degree / norm --------------------------------

__global__ __launch_bounds__(256) void deg_init(float* __restrict__ deg, int n) {
    int i = blockIdx.x * blockDim.x + threadIdx.x;
    if (i < n) deg[i] = 1.0f;                     // self-loop contribution
}

__global__ __launch_bounds__(256) void deg_edges(const int* __restrict__ ei,
                                                 float* __restrict__ deg, int E) {
    int e = blockIdx.x * blockDim.x + threadIdx.x;
    if (e < E) atomicAdd(&deg[ei[(size_t)E + e]], 1.0f);   // deg over dst
}

__global__ __launch_bounds__(256) void deg_to_dinv(float* __restrict__ d, int n) {
    int i = blockIdx.x * blockDim.x + threadIdx.x;
    if (i < n) d[i] = rsqrtf(d[i]);               // deg >= 1 always (self-loop)
}

// ------------------------------- GEMM 1 ------------------------------------
// g1[i,c] = dinv[i] * sum_k x[i,k]*W1[k,c], c in [0,64). Also seeds acc1.
// One wave = 16 output rows x 64 cols (4 WMMA accumulators), K=256 in steps of 4.

__global__ __launch_bounds__(128)
void gemm1_wmma(const float* __restrict__ x, const float* __restrict__ W1,
                const float* __restrict__ dinv,
                float* __restrict__ g1, float* __restrict__ acc1, int nrows) {
    const int wave = threadIdx.x >> 5;
    const int lane = threadIdx.x & 31;
    const int m0   = (blockIdx.x * 4 + wave) * 16;
    if (m0 >= nrows) return;                      // wave-uniform exit
    const int half = lane >> 4;                   // 0: K+{0,1}, 1: K+{2,3}
    const int r    = lane & 15;

    int arow = m0 + r; if (arow >= nrows) arow = nrows - 1;   // keep EXEC all-1s
    const float* xrow = x + (size_t)arow * IN_C;

    v8f c0 = {}, c1 = {}, c2 = {}, c3 = {};
    for (int k = 0; k < IN_C; k += 4) {
        v2f a = *(const v2f*)(xrow + k + 2 * half);           // A 16x4 fragment
        const float* w0 = W1 + (size_t)(k + 2 * half) * HID_C;
        const float* w1 = w0 + HID_C;
        v2f b0 = { w0[r],      w1[r]      };
        v2f b1 = { w0[r + 16], w1[r + 16] };
        v2f b2 = { w0[r + 32], w1[r + 32] };
        v2f b3 = { w0[r + 48], w1[r + 48] };
        c0 = __builtin_amdgcn_wmma_f32_16x16x4_f32(false, a, false, b0, (short)0, c0, false, false);
        c1 = __builtin_amdgcn_wmma_f32_16x16x4_f32(false, a, false, b1, (short)0, c1, false, false);
        c2 = __builtin_amdgcn_wmma_f32_16x16x4_f32(false, a, false, b2, (short)0, c2, false, false);
        c3 = __builtin_amdgcn_wmma_f32_16x16x4_f32(false, a, false, b3, (short)0, c3, false, false);
    }
#pragma unroll
    for (int v = 0; v < 8; ++v) {
        int row = m0 + v + 8 * half;              // C/D layout: M = v (+8 hi half)
        if (row < nrows) {
            float s = dinv[row];
            size_t base = (size_t)row * HID_C + r;
            float o;
            o = s * c0[v]; g1[base     ] = o; acc1[base     ] = o;
            o = s * c1[v]; g1[base + 16] = o; acc1[base + 16] = o;
            o = s * c2[v]; g1[base + 32] = o; acc1[base + 32] = o;
            o = s * c3[v]; g1[base + 48] = o; acc1[base + 48] = o;
        }
    }
}

// ------------------------------- GEMM 2 ------------------------------------
// g2[i,c] = dinv[i] * sum_k h[i,k]*W2[k,c], c in [0,40): 3 col tiles, last guarded.

__global__ __launch_bounds__(128)
void gemm2_wmma(const float* __restrict__ h, const float* __restrict__ W2,
                const float* __restrict__ dinv,
                float* __restrict__ g2, int nrows) {
    const int wave = threadIdx.x >> 5;
    const int lane = threadIdx.x & 31;
    const int m0   = (blockIdx.x * 4 + wave) * 16;
    if (m0 >= nrows) return;
    const int half = lane >> 4;
    const int r    = lane & 15;

    int arow = m0 + r; if (arow >= nrows) arow = nrows - 1;
    const float* hrow = h + (size_t)arow * HID_C;
    const bool tail = (r + 32) < OUT_C;           // cols 32..39 valid only for r<8

    v8f c0 = {}, c1 = {}, c2 = {};
    for (int k = 0; k < HID_C; k += 4) {
        v2f a = *(const v2f*)(hrow + k + 2 * half);
        const float* w0 = W2 + (size_t)(k + 2 * half) * OUT_C;
        const float* w1 = w0 + OUT_C;
        v2f b0 = { w0[r],      w1[r]      };
        v2f b1 = { w0[r + 16], w1[r + 16] };
        v2f b2 = { tail ? w0[r + 32] : 0.0f, tail ? w1[r + 32] : 0.0f };
        c0 = __builtin_amdgcn_wmma_f32_16x16x4_f32(false, a, false, b0, (short)0, c0, false, false);
        c1 = __builtin_amdgcn_wmma_f32_16x16x4_f32(false, a, false, b1, (short)0, c1, false, false);
        c2 = __builtin_amdgcn_wmma_f32_16x16x4_f32(false, a, false, b2, (short)0, c2, false, false);
    }
#pragma unroll
    for (int v = 0; v < 8; ++v) {
        int row = m0 + v + 8 * half;
        if (row < nrows) {
            float s = dinv[row];
            size_t base = (size_t)row * OUT_C + r;
            g2[base     ] = s * c0[v];
            g2[base + 16] = s * c1[v];
            if (tail) g2[base + 32] = s * c2[v];
        }
    }
}

// ------------------------------ scatter ------------------------------------
// Wave-per-edge: acc[dst,:] += g[src,:]. Coalesced loads; f32 atomics land in
// L2 (acc buffers fit in the 192MB L2, so this phase is L2-resident).

__global__ __launch_bounds__(256)
void scatter64(const int* __restrict__ ei, const float* __restrict__ g,
               float* __restrict__ acc, int E) {
    int e    = (int)((blockIdx.x * (unsigned)blockDim.x + threadIdx.x) >> 5);
    int lane = threadIdx.x & 31;
    if (e >= E) return;
    int src = ei[e], dst = ei[(size_t)E + e];
    const float* gs = g   + (size_t)src * HID_C;
    float*       ad = acc + (size_t)dst * HID_C;
    atomicAdd(ad + lane,      gs[lane]);
    atomicAdd(ad + lane + 32, gs[lane + 32]);
}

__global__ __launch_bounds__(256)
void scatter40(const int* __restrict__ ei, const float* __restrict__ g,
               float* __restrict__ acc, int E) {
    int e    = (int)((blockIdx.x * (unsigned)blockDim.x + threadIdx.x) >> 5);
    int lane = threadIdx.x & 31;
    if (e >= E) return;
    int src = ei[e], dst = ei[(size_t)E + e];
    const float* gs = g   + (size_t)src * OUT_C;
    float*       ad = acc + (size_t)dst * OUT_C;
    atomicAdd(ad + lane, gs[lane]);
    if (lane < OUT_C - 32) atomicAdd(ad + lane + 32, gs[lane + 32]);
}

// ------------------------------ finalize -----------------------------------

__global__ __launch_bounds__(256)
void finalize1(const float* __restrict__ acc1, const float* __restrict__ dinv,
               const float* __restrict__ b1, float* __restrict__ h, int n) {
    int i = blockIdx.x * blockDim.x + threadIdx.x;
    if (i < n) {
        int row = i >> 6, c = i & 63;
        float v = dinv[row] * acc1[i] + b1[c];
        h[i] = fmaxf(v, 0.0f);                    // relu
    }
}

__global__ __launch_bounds__(256)
void finalize2(const float* __restrict__ acc2, const float* __restrict__ dinv,
               const float* __restrict__ b2, float* __restrict__ out, int n) {
    int i = blockIdx.x * blockDim.x + threadIdx.x;
    if (i < n) {
        int row = i / OUT_C, c = i - row * OUT_C;
        out[i] = dinv[row] * acc2[i] + b2[c];
    }
}

__global__ __launch_bounds__(256)
void copy_f32(const float* __restrict__ src, float* __restrict__ dst, int n) {
    int i = blockIdx.x * blockDim.x + threadIdx.x;
    if (i < n) dst[i] = src[i];
}

// ------------------------------ launcher -----------------------------------

extern "C" void kernel_launch(void* const* d_in, const int* in_sizes, int n_in,
                              void* d_out, int out_size, void* d_ws, size_t ws_size,
                              hipStream_t stream) {
    const float* x  = (const float*)d_in[0];
    const int*   ei = (const int*)  d_in[1];
    const float* W1 = (const float*)d_in[2];
    const float* b1 = (const float*)d_in[3];
    const float* W2 = (const float*)d_in[4];
    const float* b2 = (const float*)d_in[5];
    float* out = (float*)d_out;

    const int N = in_sizes[0] / IN_C;
    const int E = in_sizes[1] / 2;

    // Workspace layout (~52MB): dinv | buf1 (g1 -> h -> acc2) | buf2 (acc1 -> g2)
    float* dinv = (float*)d_ws;
    size_t nAlign = (((size_t)N) + 127) & ~(size_t)127;
    float* buf1 = dinv + nAlign;
    float* buf2 = buf1 + (size_t)N * HID_C;

    const int tiles   = (N + 15) / 16;
    const int gemmGrd = (tiles + 3) / 4;
    const int edgeGrd = (E + 7) / 8;              // 8 waves (edges) per 256-thr block

    deg_init  <<<(N + 255) / 256, 256, 0, stream>>>(dinv, N);
    deg_edges <<<(E + 255) / 256, 256, 0, stream>>>(ei, dinv, E);
    deg_to_dinv<<<(N + 255) / 256, 256, 0, stream>>>(dinv, N);

    // Layer 1: g1=buf1, acc1=buf2 (seeded with self-loop message inside GEMM)
    gemm1_wmma<<<gemmGrd, 128, 0, stream>>>(x, W1, dinv, buf1, buf2, N);
    scatter64 <<<edgeGrd, 256, 0, stream>>>(ei, buf1, buf2, E);
    finalize1 <<<(N * HID_C + 255) / 256, 256, 0, stream>>>(buf2, dinv, b1, buf1, N * HID_C);
    // buf1 now holds h (g1 dead), buf2 free

    // Layer 2: g2=buf2; then acc2=buf1 (h dead after gemm2 completes)
    gemm2_wmma<<<gemmGrd, 128, 0, stream>>>(buf1, W2, dinv, buf2, N);
    copy_f32  <<<(N * OUT_C + 255) / 256, 256, 0, stream>>>(buf2, buf1, N * OUT_C);
    scatter40 <<<edgeGrd, 256, 0, stream>>>(ei, buf2, buf1, E);
    finalize2 <<<(N * OUT_C + 255) / 256, 256, 0, stream>>>(buf1, dinv, b2, out, N * OUT_C);
}